// Net_18348100288587
// MI455X (gfx1250) — compile-verified
//
#include <hip/hip_runtime.h>

typedef __attribute__((ext_vector_type(2))) float v2f;
typedef __attribute__((ext_vector_type(8))) float v8f;

// ---------------- degree / normalization ----------------

__global__ void k_init_deg(float* __restrict__ deg, int n) {
    int i = blockIdx.x * blockDim.x + threadIdx.x;
    if (i < n) deg[i] = 1.0f;                 // self-loop contributes 1
}

__global__ void k_deg_edges(const int* __restrict__ ei, float* __restrict__ deg, int E) {
    int e = blockIdx.x * blockDim.x + threadIdx.x;
    if (e < E) atomicAdd(&deg[ei[E + e]], 1.0f);   // dst = ei[1][e]
}

__global__ void k_rsqrt(float* __restrict__ deg, int n) {
    int i = blockIdx.x * blockDim.x + threadIdx.x;
    if (i < n) deg[i] = 1.0f / sqrtf(deg[i]);      // deg >= 1 always
}

// ---------------- fp32 WMMA GEMM: C[M x N] = A[M x K] @ B[K x N], K = 128 ----------------
// One wave owns a 16-row strip and all N/16 column tiles.
// B is staged into LDS *pair-swizzled*: the float2 (B[k][n], B[k+1][n]) lives at
// pair-index (k/2)*N + n, so every WMMA B-fragment is one aligned 8-byte DS load
// (compiler pairs them into ds_load_2addr_b64) addressed as base + immediate.
// Per k-step, all NT fragments are fetched into distinct register quads before the
// WMMA burst so DS latency is pipelined instead of serialized on one quad.

template <int N>
__global__ void k_gemm_wmma(const float* __restrict__ A, const float* __restrict__ B,
                            float* __restrict__ C, int M) {
    constexpr int K  = 128;
    constexpr int NT = N / 16;                 // 8 (N=128) or 4 (N=64)
    __shared__ v2f sB[(K / 2) * N];            // 64KB or 32KB

    const int tid = threadIdx.x;
    for (int i = tid * 4; i < K * N; i += blockDim.x * 4) {
        const float4 bv = *(const float4*)(B + i);
        const int k  = i / N;
        const int n  = i % N;                  // n..n+3 all < N (4 | N)
        const int kp = (k >> 1) * N + n;
        const int kb = k & 1;
        sB[kp + 0][kb] = bv.x;
        sB[kp + 1][kb] = bv.y;
        sB[kp + 2][kb] = bv.z;
        sB[kp + 3][kb] = bv.w;
    }
    __syncthreads();

    const int wave = tid >> 5;                 // 8 waves per block of 256
    const int lane = tid & 31;
    const int m0 = (blockIdx.x * 8 + wave) * 16;
    if (m0 >= M) return;                       // M % 16 == 0 -> whole-wave uniform exit

    const int mr = lane & 15;                  // row within tile / column within tile
    const int kh = lane >> 4;                  // 0: K=k,k+1   1: K=k+2,k+3

    v8f acc[NT];
    #pragma unroll
    for (int j = 0; j < NT; ++j) acc[j] = v8f{0.f,0.f,0.f,0.f,0.f,0.f,0.f,0.f};

    const float* arow  = A + (size_t)(m0 + mr) * K;
    const v2f*   bbase = sB + kh * N + mr;     // per-lane base; rest is immediate offsets

    #pragma unroll 4
    for (int k0 = 0; k0 < K; k0 += 4) {
        const v2f a = *(const v2f*)(arow + k0 + 2 * kh);
        const int prow = (k0 >> 1) * N;
        v2f bf[NT];
        #pragma unroll
        for (int j = 0; j < NT; ++j) bf[j] = bbase[prow + j * 16];
        #pragma unroll
        for (int j = 0; j < NT; ++j) {
            acc[j] = __builtin_amdgcn_wmma_f32_16x16x4_f32(
                false, a, false, bf[j], (short)0, acc[j], false, false);
        }
    }

    // C/D layout: VGPR v, lanes 0-15 -> M=v, N=lane; lanes 16-31 -> M=v+8, N=lane-16
    #pragma unroll
    for (int j = 0; j < NT; ++j) {
        #pragma unroll
        for (int v = 0; v < 8; ++v) {
            C[(size_t)(m0 + v + 8 * kh) * N + j * 16 + mr] = acc[j][v];
        }
    }
}

// ---------------- edge scatter-add: agg[dst] += h[src] * dis[src] * dis[dst] ----------------
// One wave per edge; C/32 floats per lane. Feature tables (<=51MB) are L2-resident.

template <int C>
__global__ void k_edge_agg(const int* __restrict__ ei, const float* __restrict__ dis,
                           const float* __restrict__ h, float* __restrict__ agg, int E) {
    static_assert(C == 128 || C == 64, "channel count");
    const int gid  = blockIdx.x * blockDim.x + threadIdx.x;
    const int e    = gid >> 5;
    const int lane = gid & 31;
    if (e >= E) return;
    const int s = ei[e];
    const int d = ei[E + e];
    const float w = dis[s] * dis[d];
    constexpr int V = C / 32;
    const float* hp = h   + (size_t)s * C + lane * V;
    float*       ap = agg + (size_t)d * C + lane * V;
    if constexpr (V == 4) {
        const float4 hv = *(const float4*)hp;
        atomicAdd(ap + 0, hv.x * w);
        atomicAdd(ap + 1, hv.y * w);
        atomicAdd(ap + 2, hv.z * w);
        atomicAdd(ap + 3, hv.w * w);
    } else {
        const float2 hv = *(const float2*)hp;
        atomicAdd(ap + 0, hv.x * w);
        atomicAdd(ap + 1, hv.y * w);
    }
}

// ---------------- pointwise ----------------

template <int C>
__global__ void k_selfloop(const float* __restrict__ h, const float* __restrict__ dis,
                           float* __restrict__ agg, long long total) {
    long long i = (long long)blockIdx.x * blockDim.x + threadIdx.x;
    if (i >= total) return;
    const float di = dis[i / C];
    agg[i] = h[i] * di * di;
}

template <int C>
__global__ void k_bias_relu(const float* __restrict__ agg, const float* __restrict__ b,
                            float* __restrict__ out, long long total) {
    long long i = (long long)blockIdx.x * blockDim.x + threadIdx.x;
    if (i >= total) return;
    out[i] = fmaxf(agg[i] + b[i % C], 0.0f);
}

template <int C>
__global__ void k_selfloop_bias(const float* __restrict__ h, const float* __restrict__ dis,
                                const float* __restrict__ b, float* __restrict__ out,
                                long long total) {
    long long i = (long long)blockIdx.x * blockDim.x + threadIdx.x;
    if (i >= total) return;
    const float di = dis[i / C];
    out[i] = h[i] * di * di + b[i % C];
}

// ---------------- launch ----------------

extern "C" void kernel_launch(void* const* d_in, const int* in_sizes, int n_in,
                              void* d_out, int out_size, void* d_ws, size_t ws_size,
                              hipStream_t stream) {
    const float* x  = (const float*)d_in[0];
    const int*   ei = (const int*)d_in[1];
    const float* W1 = (const float*)d_in[2];
    const float* b1 = (const float*)d_in[3];
    const float* W2 = (const float*)d_in[4];
    const float* b2 = (const float*)d_in[5];
    float*       out = (float*)d_out;

    const int N = in_sizes[0] / 128;   // 100000
    const int E = in_sizes[1] / 2;     // 1600000

    char* ws = (char*)d_ws;
    size_t off = 0;
    float* dis  = (float*)(ws + off); off += (((size_t)N * 4) + 255) & ~(size_t)255;
    float* bufA = (float*)(ws + off); off += (size_t)N * 128 * 4;   // h1, then relu(h)
    float* bufB = (float*)(ws + off); off += (size_t)N * 128 * 4;   // agg1, then h2

    const int T = 256;
    // --- normalization ---
    k_init_deg <<<(N + T - 1) / T, T, 0, stream>>>(dis, N);
    k_deg_edges<<<(E + T - 1) / T, T, 0, stream>>>(ei, dis, E);
    k_rsqrt    <<<(N + T - 1) / T, T, 0, stream>>>(dis, N);

    // --- layer 1 ---
    k_gemm_wmma<128><<<(N + 127) / 128, T, 0, stream>>>(x, W1, bufA, N);     // h1 = x @ W1
    {
        long long tot = (long long)N * 128;
        k_selfloop<128><<<(unsigned)((tot + T - 1) / T), T, 0, stream>>>(bufA, dis, bufB, tot);
    }
    k_edge_agg<128><<<(E + 7) / 8, T, 0, stream>>>(ei, dis, bufA, bufB, E);  // scatter-add
    {
        long long tot = (long long)N * 128;
        k_bias_relu<128><<<(unsigned)((tot + T - 1) / T), T, 0, stream>>>(bufB, b1, bufA, tot);
    }

    // --- layer 2 ---
    k_gemm_wmma<64><<<(N + 127) / 128, T, 0, stream>>>(bufA, W2, bufB, N);   // h2 = h @ W2
    {
        long long tot = (long long)N * 64;
        k_selfloop_bias<64><<<(unsigned)((tot + T - 1) / T), T, 0, stream>>>(bufB, dis, b2, out, tot);
    }
    k_edge_agg<64><<<(E + 7) / 8, T, 0, stream>>>(ei, dis, bufB, out, E);
}